// PerturbedTopK_51127290692284
// MI455X (gfx1250) — compile-verified
//
#include <hip/hip_runtime.h>

// Problem constants (fixed by the reference)
#define B_    32
#define N_    250
#define D_    1024
#define K_    102          // int(1024 * 0.1)
#define CHUNK 5            // samples per workgroup (double-buffered async staging)
#define NT    256          // threads per block (8 waves of 32 on CDNA5)
#define NW    (NT / 32)    // waves per block
#define EPT   4            // elements (d's) per thread: 256*4 = 1024

__constant__ const float kSigma = 0.05f;
__constant__ const float kInvN  = 1.0f / 250.0f;

#if __has_builtin(__builtin_amdgcn_global_load_async_to_lds_b128)
#define HAS_ASYNC_B128 1
#endif

// Builtin parameter types: (v4i addrspace(1)*, v4i addrspace(3)*, imm i32, imm i32)
typedef int v4i __attribute__((vector_size(4 * sizeof(int))));
typedef __attribute__((address_space(1))) v4i* gv4i_p;  // global
typedef __attribute__((address_space(3))) v4i* lv4i_p;  // LDS

// Copy 16 bytes global -> LDS. Async (ASYNCcnt path) when available.
__device__ __forceinline__ void async_copy16(const float* g, float* l) {
#ifdef HAS_ASYNC_B128
    __builtin_amdgcn_global_load_async_to_lds_b128((gv4i_p)g, (lv4i_p)l, 0, 0);
#else
    float4 v = *(const float4*)g;
    *(float4*)l = v;
#endif
}

template <int N>
__device__ __forceinline__ void wait_async() {
#if __has_builtin(__builtin_amdgcn_s_wait_asynccnt)
    __builtin_amdgcn_s_wait_asynccnt(N);
#elif defined(HAS_ASYNC_B128)
    asm volatile("s_wait_asynccnt %0" ::"n"(N));
#endif
}

// Order-preserving float -> uint32 map (larger float => larger uint)
__device__ __forceinline__ unsigned f2ord(float f) {
    unsigned u = __float_as_uint(f);
    return u ^ ((u & 0x80000000u) ? 0xFFFFFFFFu : 0x80000000u);
}

// Intra-wave (32-lane) inclusive SUFFIX scan: v[l] = sum_{j>=l} in[j]
__device__ __forceinline__ unsigned wave_suffix_incl(unsigned v, int lane) {
#pragma unroll
    for (int off = 1; off < 32; off <<= 1) {
        unsigned t = __shfl_down((int)v, off, 32);
        if (lane + off < 32) v += t;
    }
    return v;
}

// Intra-wave (32-lane) inclusive PREFIX scan: v[l] = sum_{j<=l} in[j]
__device__ __forceinline__ unsigned wave_prefix_incl(unsigned v, int lane) {
#pragma unroll
    for (int off = 1; off < 32; off <<= 1) {
        unsigned t = __shfl_up((int)v, off, 32);
        if (lane >= off) v += t;
    }
    return v;
}

__global__ void zero_f32(float* __restrict__ out, int n) {
    int i = blockIdx.x * blockDim.x + threadIdx.x;
    if (i < n) out[i] = 0.0f;
}

__global__ void __launch_bounds__(NT)
perturbed_topk_kernel(const float* __restrict__ x,
                      const float* __restrict__ noise,
                      float* __restrict__ out) {
    __shared__ __align__(16) float    s_x[D_];          // 4 KB, reused across chunk
    __shared__ __align__(16) float    s_noise[2][D_];   // 8 KB double buffer
    __shared__ unsigned               s_hist[NT];       // 256-bin histogram
    __shared__ unsigned               s_wtot[NW];       // per-wave scan totals
    __shared__ unsigned               s_bin;            // broadcast: selected radix bin
    __shared__ unsigned               s_above;          // broadcast: count strictly above bin

    const int tid     = threadIdx.x;
    const int lane    = tid & 31;
    const int wid     = tid >> 5;
    const int nchunks = N_ / CHUNK;
    const int b       = blockIdx.x / nchunks;
    const int s0      = (blockIdx.x % nchunks) * CHUNK;

    // Stage x row (visible after the first barrier below)
    {
        float4 xv = *(const float4*)(x + (size_t)b * D_ + tid * EPT);
        *(float4*)(&s_x[tid * EPT]) = xv;
    }

    const float* nbase = noise + ((size_t)b * N_ + s0) * D_;

    // Prime the pipeline: async-stage sample 0
    async_copy16(nbase + tid * EPT, &s_noise[0][tid * EPT]);

    for (int i = 0; i < CHUNK; ++i) {
        // Issue next sample's DMA while we still compute on the current one
        if (i + 1 < CHUNK) {
            async_copy16(nbase + (size_t)(i + 1) * D_ + tid * EPT,
                         &s_noise[(i + 1) & 1][tid * EPT]);
            wait_async<1>();   // oldest (sample i) done; sample i+1 may remain in flight
        } else {
            wait_async<0>();
        }
        __syncthreads();       // all waves' sample-i slices landed in LDS

        // ---- load current sample, compute order keys ----
        const float4 xv = *(const float4*)(&s_x[tid * EPT]);
        const float4 nv = *(const float4*)(&s_noise[i & 1][tid * EPT]);
        float p[EPT] = { xv.x + kSigma * nv.x, xv.y + kSigma * nv.y,
                         xv.z + kSigma * nv.z, xv.w + kSigma * nv.w };
        unsigned u[EPT];
#pragma unroll
        for (int j = 0; j < EPT; ++j) u[j] = f2ord(p[j]);

        // ---- 4-pass radix select: exact k-th largest key T, tie budget ----
        unsigned prefix = 0u, mask = 0u, krem = K_;
#pragma unroll
        for (int pass = 0; pass < 4; ++pass) {
            const int shift = 24 - 8 * pass;
            s_hist[tid] = 0u;
            __syncthreads();                                   // (a) zero visible
#pragma unroll
            for (int j = 0; j < EPT; ++j)
                if ((u[j] & mask) == prefix)
                    atomicAdd(&s_hist[(u[j] >> shift) & 0xFFu], 1u);
            __syncthreads();                                   // (b) histogram complete

            // suffix count for own bin: intra-wave shuffle scan + 8-wave combine
            const unsigned cnt = s_hist[tid];
            unsigned v = wave_suffix_incl(cnt, lane);          // no barriers
            if (lane == 0) s_wtot[wid] = v;                    // wave total (suffix@lane0)
            __syncthreads();                                   // (c) wave totals visible
            unsigned base = 0u;
#pragma unroll
            for (int w = 0; w < NW; ++w)
                if (w > wid) base += s_wtot[w];
            const unsigned sfx  = v + base;                    // #elems with byte >= tid
            const unsigned sfxn = sfx - cnt;                   // #elems with byte >  tid
            if (sfx >= krem && sfxn < krem) { s_bin = (unsigned)tid; s_above = sfxn; }
            __syncthreads();                                   // (d) broadcast visible
            prefix |= (s_bin << shift);
            mask   |= (0xFFu << shift);
            krem   -= s_above;
        }
        const unsigned T     = prefix;  // k-th largest key
        const unsigned t_inc = krem;    // #elements == T to include (lowest indices first)

        // ---- packed (greater, equal) exclusive scan over d = 0..1023 ----
        unsigned c[EPT], loc[EPT], run = 0u;
#pragma unroll
        for (int j = 0; j < EPT; ++j) {
            unsigned g = (u[j] > T) ? 1u : 0u;
            unsigned e = (u[j] == T) ? 1u : 0u;
            c[j]  = g | (e << 16);
            loc[j] = run;
            run   += c[j];
        }
        unsigned inc = wave_prefix_incl(run, lane);            // intra-wave inclusive
        if (lane == 31) s_wtot[wid] = inc;                     // wave total
        __syncthreads();                                       // (e) wave totals visible
        unsigned wbase = 0u;
#pragma unroll
        for (int w = 0; w < NW; ++w)
            if (w < wid) wbase += s_wtot[w];
        const unsigned excl = wbase + inc - run;               // exclusive prefix @ first elem

        // ---- emit: rank j = gpre + min(epre, t_inc); out[b, j, d] += 1/n ----
#pragma unroll
        for (int j = 0; j < EPT; ++j) {
            unsigned P    = excl + loc[j];
            unsigned gpre = P & 0xFFFFu;
            unsigned epre = P >> 16;
            bool     g    = (u[j] > T);
            bool     e    = (u[j] == T);
            if (g || (e && epre < t_inc)) {
                unsigned rank = gpre + (epre < t_inc ? epre : t_inc);
                int      d    = tid * EPT + j;
                atomicAdd(out + ((size_t)(b * K_ + (int)rank)) * D_ + d, kInvN);
            }
        }
        __syncthreads();  // (f) all reads of s_noise[i&1] done before it is re-targeted
    }
}

extern "C" void kernel_launch(void* const* d_in, const int* in_sizes, int n_in,
                              void* d_out, int out_size, void* d_ws, size_t ws_size,
                              hipStream_t stream) {
    const float* x     = (const float*)d_in[0];   // (32, 1024) fp32
    const float* noise = (const float*)d_in[1];   // (32, 250, 1024) fp32
    float*       out   = (float*)d_out;           // (32, 102, 1024) fp32

    zero_f32<<<(out_size + NT - 1) / NT, NT, 0, stream>>>(out, out_size);
    perturbed_topk_kernel<<<B_ * (N_ / CHUNK), NT, 0, stream>>>(x, noise, out);
}